// FrictionlessFuruta_82085414961423
// MI455X (gfx1250) — compile-verified
//
#include <hip/hip_runtime.h>

// Furuta pendulum frictionless dynamics, explicit Euler (semi-implicit on pos).
// Memory-bound streaming map: 36 B per row (~650 Grow/s at 23.3 TB/s).
// Keep VALU/row low enough to stay under the HBM roof: native v_sin/v_cos
// (no libm Payne-Hanek slow path) and v_rcp + 1 Newton step for the divide.

// Native clang vector so __builtin_nontemporal_store accepts it.
typedef __attribute__((ext_vector_type(4))) float v4f;

#define B_DT   0.01f
#define B_G    9.8f
#define B_MP   0.024f
#define B_LR   0.085f
#define B_LP   0.129f
#define B_RM   8.4f
#define B_KT   0.042f
#define B_KM   0.042f
#define B_J1   0.0002306f
#define B_J2   0.00013313f

#define INV_2PI 0.15915494309189535f   // 1/(2*pi): v_sin/v_cos take revolutions

__global__ __launch_bounds__(256) void
furuta_step_kernel(const v4f* __restrict__ state,
                   const float* __restrict__ action,
                   v4f*       __restrict__ out,
                   int n)
{
    const int i = blockIdx.x * blockDim.x + threadIdx.x;
    if (i >= n) return;

    // One full row per lane: global_load_b128 (coalesced, RT hint so the
    // 168MB input set stays resident in the 192MB L2 across graph replays).
    const v4f   s = state[i];
    const float a = action[i];

    const float beta = s.y;
    const float da   = s.z;
    const float db   = s.w;

    // Native transcendental unit: single v_sin_f32 / v_cos_f32, no branches.
    // Hardware expects the angle in revolutions (radians * 1/2pi); inputs are
    // standard-normal so |beta| is far inside the accurate range.
    const float t  = beta * INV_2PI;
    const float sb = __builtin_amdgcn_sinf(t);
    const float cb = __builtin_amdgcn_cosf(t);

    // Compile-time constant folds:
    const float c_qLP2  = 0.25f * B_MP * B_LP * B_LP;   // 0.25*MP*LP^2
    const float c_hLRLP = 0.5f  * B_MP * B_LR * B_LP;   // 0.5*MP*LR*LP
    const float c_hLP   = 0.5f  * B_MP * B_LP;          // 0.5*MP*LP
    const float c_g2    = 0.5f  * B_MP * B_G  * B_LP;   // 0.5*MP*G*LP
    const float c_tau_a = -5.0f * B_KT / B_RM;          // KT*(-5)/RM
    const float c_tau_d = -(B_KT * B_KM) / B_RM;        // KT*(-KM)/RM

    // Mass matrix (m22 == J2 constant)
    const float m11 = B_J1 + c_qLP2 * sb * sb;
    const float m12 = c_hLRLP * cb;

    // Coriolis / gravity / input torque
    const float n1  = c_hLP * sb * (B_LP * da * db * cb - B_LR * db * db);
    const float n2  = c_hLP * sb * (-0.5f * B_LP * da * da * cb);
    const float g2  = c_g2 * sb;
    const float tau = c_tau_a * a + c_tau_d * da;

    const float r1 = tau - n1;
    const float r2 = -n2 - g2;

    // Solve 2x2 system: v_rcp_f32 + one Newton-Raphson refinement
    // (3 VALU ops instead of the ~12-op IEEE div_scale/div_fmas sequence).
    const float det = m11 * B_J2 - m12 * m12;
    float rdet = __builtin_amdgcn_rcpf(det);
    rdet = rdet * (2.0f - det * rdet);
    const float dd1 = (B_J2 * r1 - m12 * r2) * rdet;
    const float dd2 = (m11 * r2 - m12 * r1) * rdet;

    // Semi-implicit Euler: new_vel then new_pos from new_vel
    const float nv0 = da  + dd1 * B_DT;
    const float nv1 = db  + dd2 * B_DT;
    const float np0 = s.x + nv0 * B_DT;
    const float np1 = s.y + nv1 * B_DT;

    // Write-once streaming output: non-temporal 128-bit store (TH=NT) so the
    // 128MB output doesn't evict the input working set from L2 between replays.
    v4f res;
    res.x = np0; res.y = np1; res.z = nv0; res.w = nv1;
    __builtin_nontemporal_store(res, &out[i]);
}

extern "C" void kernel_launch(void* const* d_in, const int* in_sizes, int n_in,
                              void* d_out, int out_size, void* d_ws, size_t ws_size,
                              hipStream_t stream)
{
    (void)n_in; (void)d_ws; (void)ws_size; (void)out_size;

    const v4f*   state  = (const v4f*)d_in[0];   // (B,4) f32, row-major
    const float* action = (const float*)d_in[1]; // (B,1) f32
    v4f*         out    = (v4f*)d_out;           // (B,4) f32

    const int n = in_sizes[0] / 4;               // rows
    const int block = 256;                       // 8 wave32 per block
    const int grid  = (n + block - 1) / block;

    furuta_step_kernel<<<grid, block, 0, stream>>>(state, action, out, n);
}